// CorrelationModule_32890859553291
// MI455X (gfx1250) — compile-verified
//
#include <hip/hip_runtime.h>

// Problem constants (from reference): B=1, C=256, H=48, W=96, R=4, D=9.
#define C_DIM 256
#define H_DIM 48
#define W_DIM 96
#define HW    4608          // H*W
#define KPAD  84            // 81 DAP inputs padded to 21 k-steps of 4 (f32 WMMA K=4)
#define MPAD  96            // 81 DAP outputs padded to 6 M-tiles of 16
#define NPANEL 64           // N columns per workgroup in the GEMM (4 N-tiles)

typedef float v2f __attribute__((ext_vector_type(2)));
typedef float v8f __attribute__((ext_vector_type(8)));
typedef unsigned int u32x4 __attribute__((ext_vector_type(4)));
typedef int i32x4 __attribute__((ext_vector_type(4)));
typedef int i32x8 __attribute__((ext_vector_type(8)));

// ---------------------------------------------------------------------------
// Kernel 0: transpose f1/f2 from (C, H*W) to (H*W, C) so per-pixel channel
// vectors are contiguous (coalesced 32B/lane dot-product loads later).
// grid: (HW/32=144, C/32=8, 2[f1,f2]), block: (32, 8)
// ---------------------------------------------------------------------------
__global__ __launch_bounds__(256) void transpose_chw_hwc(
    const float* __restrict__ src0, const float* __restrict__ src1,
    float* __restrict__ dst0, float* __restrict__ dst1) {
  __shared__ float tile[32][33];
  const float* src = blockIdx.z ? src1 : src0;
  float*       dst = blockIdx.z ? dst1 : dst0;
  const int p0 = blockIdx.x * 32;   // pixel tile base
  const int c0 = blockIdx.y * 32;   // channel tile base
  const int tx = threadIdx.x, ty = threadIdx.y;
#pragma unroll
  for (int i = 0; i < 4; ++i) {
    const int c = c0 + ty + i * 8;
    tile[ty + i * 8][tx] = src[(size_t)c * HW + p0 + tx];
  }
  __syncthreads();
#pragma unroll
  for (int i = 0; i < 4; ++i) {
    const int p = p0 + ty + i * 8;
    dst[(size_t)p * C_DIM + c0 + tx] = tile[tx][ty + i * 8];
  }
}

// ---------------------------------------------------------------------------
// Kernel 1: correlation. One wave32 per pixel. Displacements are integer, so
// all 81 lookups share one set of bilinear fractions -> compute the 10x10
// integer-offset dot grid G once, then blend 4 taps per displacement.
// Writes cf laid out [KPAD][HW] (rows 81..83 zeroed) for the WMMA GEMM.
// grid: HW/8 = 576 blocks of 256 threads (8 waves).
// ---------------------------------------------------------------------------
__global__ __launch_bounds__(256) void corr_kernel(
    const float* __restrict__ f1t, const float* __restrict__ f2t,
    const float* __restrict__ coords, float* __restrict__ cf) {
  __shared__ float Gs[8][104];      // 10x10 grid per wave (padded)
  const int wave = threadIdx.x >> 5;
  const int lane = threadIdx.x & 31;
  const int p = blockIdx.x * 8 + wave;

  const float cx = coords[p];
  const float cy = coords[HW + p];
  const float fx0 = floorf(cx), fy0 = floorf(cy);
  const float wx1 = cx - fx0, wx0 = 1.0f - wx1;
  const float wy1 = cy - fy0, wy0 = 1.0f - wy1;
  const int ix0 = (int)fx0, iy0 = (int)fy0;

  // f1 channel vector: lane owns channels [8*lane, 8*lane+8)
  const float* f1p = f1t + (size_t)p * C_DIM + lane * 8;
  const float4 f1a = *(const float4*)f1p;
  const float4 f1b = *(const float4*)(f1p + 4);

  for (int dy = 0; dy < 10; ++dy) {
    const int y = iy0 - 4 + dy;
    const bool yok = (y >= 0) && (y < H_DIM);
    for (int dx = 0; dx < 10; ++dx) {
      const int x = ix0 - 4 + dx;
      float g = 0.0f;
      if (yok && (x >= 0) && (x < W_DIM)) {
        const float* f2p = f2t + (size_t)(y * W_DIM + x) * C_DIM + lane * 8;
        const float4 a = *(const float4*)f2p;
        const float4 b = *(const float4*)(f2p + 4);
        g = f1a.x * a.x + f1a.y * a.y + f1a.z * a.z + f1a.w * a.w +
            f1b.x * b.x + f1b.y * b.y + f1b.z * b.z + f1b.w * b.w;
      }
      // wave32 butterfly reduction
#pragma unroll
      for (int off = 16; off > 0; off >>= 1) g += __shfl_xor(g, off, 32);
      if (lane == 0) Gs[wave][dy * 10 + dx] = g;
    }
  }
  __syncthreads();

  const float scale = 0.0625f;  // 1/sqrt(C) = 1/16
  for (int t = lane; t < 81; t += 32) {
    const int i = t / 9;        // x-displacement index
    const int j = t % 9;        // y-displacement index
    const float g00 = Gs[wave][j * 10 + i];
    const float g10 = Gs[wave][j * 10 + i + 1];
    const float g01 = Gs[wave][(j + 1) * 10 + i];
    const float g11 = Gs[wave][(j + 1) * 10 + i + 1];
    const float v = wy0 * (wx0 * g00 + wx1 * g10) + wy1 * (wx0 * g01 + wx1 * g11);
    cf[t * HW + p] = v * scale;
  }
  if (lane < KPAD - 81) cf[(81 + lane) * HW + p] = 0.0f;  // zero K padding
}

// ---------------------------------------------------------------------------
// Kernel 2: DAP mixing out[81 x 4608] = w_dap[81x81] @ cf[81x4608] as a WMMA
// f32 GEMM (V_WMMA_F32_16X16X4_F32 keeps full f32 precision).
// Each block owns a 96(M) x 64(N) output panel:
//   - W (zero-padded 96x84) is cooperatively staged in LDS,
//   - the 84x64 B panel of cf is DMA'd into LDS by the Tensor Data Mover
//     (tensor_load_to_lds + s_wait_tensorcnt, issued by wave 0),
//   - each of the 8 waves computes 3 M-tiles for one fixed N-tile, reusing
//     its B fragment across the 3 WMMAs per k-step (63 WMMAs/wave).
// grid: HW/64 = 72 blocks of 256 threads.
// ---------------------------------------------------------------------------
__global__ __launch_bounds__(256) void dap_gemm(
    const float* __restrict__ w_dap, const float* __restrict__ cf,
    float* __restrict__ out) {
  __shared__ float Alds[MPAD * KPAD];     // 96*84*4  = 32256 B
  __shared__ float Blds[KPAD * NPANEL];   // 84*64*4  = 21504 B
  const int tid = threadIdx.x;
  const int n0 = blockIdx.x * NPANEL;

  // Stage zero-padded W into LDS.
  for (int idx = tid; idx < MPAD * KPAD; idx += 256) {
    const int m = idx / KPAD, k = idx % KPAD;
    Alds[idx] = (m < 81 && k < 81) ? w_dap[m * 81 + k] : 0.0f;
  }

#if defined(__has_builtin) && __has_builtin(__builtin_amdgcn_tensor_load_to_lds)
  // ---- Tensor Data Mover path: DMA the 84x64 f32 B panel into LDS. ----
  if (tid < 32) {
    const unsigned long long gaddr =
        (unsigned long long)(uintptr_t)cf + (unsigned long long)n0 * 4ull;
    const unsigned int lds_off = (unsigned int)(uintptr_t)&Blds[0];
    // D# group 0: count=1 | lds_addr | global_addr[56:0] | type=2
    u32x4 g0;
    g0[0] = 1u;
    g0[1] = lds_off;
    g0[2] = (unsigned int)(gaddr & 0xffffffffu);
    g0[3] = (unsigned int)((gaddr >> 32) & 0x01ffffffu) | (2u << 30);
    // D# group 1: data_size=2 (4B); tensor_dim0/1; tile_dim0/1; strides.
    const unsigned int dim0 = (unsigned int)(HW - n0);   // >= tile_dim0, no OOB
    const unsigned int dim1 = KPAD;
    const unsigned int td0  = NPANEL, td1 = KPAD;
    const unsigned long long s0 = HW;                    // dim0 stride (elems)
    const unsigned long long s1 = (unsigned long long)HW * KPAD;
    i32x8 g1;
    g1[0] = (int)(2u << 16);                                   // data_size=4B
    g1[1] = (int)((dim0 & 0xffffu) << 16);                     // dim0[15:0]@48
    g1[2] = (int)((dim0 >> 16) | ((dim1 & 0xffffu) << 16));    // dim0[31:16], dim1[15:0]
    g1[3] = (int)((dim1 >> 16) | (td0 << 16));                 // dim1[31:16], tile_dim0
    g1[4] = (int)(td1 & 0xffffu);                              // tile_dim1 (tile_dim2=0)
    g1[5] = (int)(unsigned int)(s0 & 0xffffffffu);             // stride0[31:0]
    g1[6] = (int)(unsigned int)(((s0 >> 32) & 0xffffu) |
                                ((s1 & 0xffffu) << 16));       // stride0[47:32], stride1[15:0]
    g1[7] = (int)(unsigned int)((s1 >> 16) & 0xffffffffu);     // stride1[47:16]
    const i32x4 z4 = {0, 0, 0, 0};
    const i32x8 z8 = {0, 0, 0, 0, 0, 0, 0, 0};
    // amdgpu-toolchain (clang-23) 6-arg form:
    //   (u32x4 g0, i32x8 g1, i32x4 g2, i32x4 g3, i32x8, i32 cpol)
    __builtin_amdgcn_tensor_load_to_lds(g0, g1, z4, z4, z8, 0);
    __builtin_amdgcn_s_wait_tensorcnt(0);
  }
#else
  // ---- Fallback: cooperative global->LDS staging of the B panel. ----
  for (int idx = tid; idx < KPAD * NPANEL; idx += 256) {
    const int k = idx / NPANEL, c = idx % NPANEL;
    Blds[idx] = cf[k * HW + n0 + c];
  }
#endif
  __syncthreads();

  const int wave = tid >> 5;
  const int lane = tid & 31;
  const int ntl   = wave & 3;          // N-tile within panel (0..3)
  const int mbase = (wave >> 2) * 3;   // waves 0-3 -> M-tiles 0..2, 4-7 -> 3..5
  const int col  = lane & 15;          // N position / A row position
  const int half = lane >> 4;          // second half-wave: k+2 / m+8

  v8f acc0 = {0.f, 0.f, 0.f, 0.f, 0.f, 0.f, 0.f, 0.f};
  v8f acc1 = acc0, acc2 = acc0;

  for (int kt = 0; kt < 21; ++kt) {
    const int k0 = kt * 4;
    // B 4x16 f32 fragment from LDS: VGPR0 = {K | K+2}, VGPR1 = {K+1 | K+3}
    v2f b;
    b.x = Blds[(k0 + 2 * half) * NPANEL + ntl * 16 + col];
    b.y = Blds[(k0 + 2 * half + 1) * NPANEL + ntl * 16 + col];
    // A 16x4 f32 fragments, one per M-tile; B reused across all three.
#pragma unroll
    for (int q = 0; q < 3; ++q) {
      const int arow = (mbase + q) * 16 + col;
      v2f a;
      a.x = Alds[arow * KPAD + k0 + 2 * half];
      a.y = Alds[arow * KPAD + k0 + 2 * half + 1];
      v8f& acc = (q == 0) ? acc0 : (q == 1) ? acc1 : acc2;
      acc = __builtin_amdgcn_wmma_f32_16x16x4_f32(
          /*neg_a=*/false, a, /*neg_b=*/false, b,
          /*c_mod=*/(short)0, acc, /*reuse_a=*/false, /*reuse_b=*/false);
    }
  }

  // C/D layout: VGPR r of lane l holds (M = r + 8*(l/16), N = l%16)
#pragma unroll
  for (int q = 0; q < 3; ++q) {
    const v8f& acc = (q == 0) ? acc0 : (q == 1) ? acc1 : acc2;
#pragma unroll
    for (int r = 0; r < 8; ++r) {
      const int m = (mbase + q) * 16 + r + 8 * half;
      if (m < 81) out[m * HW + n0 + ntl * 16 + col] = acc[r];
    }
  }
}

// ---------------------------------------------------------------------------
extern "C" void kernel_launch(void* const* d_in, const int* in_sizes, int n_in,
                              void* d_out, int out_size, void* d_ws, size_t ws_size,
                              hipStream_t stream) {
  const float* f1     = (const float*)d_in[0];  // (1,256,48,96)
  const float* f2     = (const float*)d_in[1];  // (1,256,48,96)
  const float* coords = (const float*)d_in[2];  // (1,2,48,96)
  const float* w_dap  = (const float*)d_in[3];  // (81,81)
  float* out = (float*)d_out;                   // (1,81,48,96)

  // Workspace layout (floats): f1t[HW*C] | f2t[HW*C] | cf[KPAD*HW]  (~10.5 MB)
  float* ws  = (float*)d_ws;
  float* f1t = ws;
  float* f2t = ws + (size_t)HW * C_DIM;
  float* cf  = ws + (size_t)2 * HW * C_DIM;

  transpose_chw_hwc<<<dim3(HW / 32, C_DIM / 32, 2), dim3(32, 8), 0, stream>>>(
      f1, f2, f1t, f2t);
  corr_kernel<<<HW / 8, 256, 0, stream>>>(f1t, f2t, coords, cf);
  dap_gemm<<<HW / NPANEL, 256, 0, stream>>>(w_dap, cf, out);
}